// KNN_6030134083767
// MI455X (gfx1250) — compile-verified
//
#include <hip/hip_runtime.h>
#include <hip/hip_bf16.h>

typedef __attribute__((ext_vector_type(2))) float v2f;
typedef __attribute__((ext_vector_type(8))) float v8f;

// Problem constants (match reference)
#define BATCH 4
#define NPTS  16384
#define MQRY  4096
#define KNN_K 16

#define CHUNK 256          // points per chunk
#define STR   20           // LDS row stride in floats (16 + 4 pad -> conflict-free)
#define FLT_BIG 3.402823466e+38f

__global__ __launch_bounds__(256) void KNN_6030134083767_kernel(
    const float* __restrict__ xyz,      // [B, N, 3]
    const float* __restrict__ new_xyz,  // [B, M, 3]
    int* __restrict__ out)              // [B, M, K]
{
    __shared__ __align__(16) float ldsD[CHUNK * STR];      // 20 KB distance staging
    __shared__ float mD[16 * 16 * KNN_K];                  // 16 KB merge dists
    __shared__ int   mI[16 * 16 * KNN_K];                  // 16 KB merge indices

    const int tid   = threadIdx.x;
    const int lane  = tid & 31;
    const int lm    = lane & 15;
    const int wv    = tid >> 5;        // wave 0..7
    const int qid   = tid & 15;        // query within tile (scan phase)
    const int slot  = tid >> 4;        // 0..15 (scan phase)

    const int wg     = blockIdx.x;
    const int mtiles = MQRY / 16;      // 256
    const int b      = wg / mtiles;
    const int mtile  = wg % mtiles;

    // ---- A operand: 16 queries, resident for whole kernel ----
    // A is 16x4 (MxK): lanes 0-15 supply K0,K1; lanes 16-31 supply K2,K3.
    // Row m: (qx, qy, qz, 1.0)
    const float* qp = new_xyz + ((size_t)b * MQRY + (size_t)mtile * 16 + lm) * 3;
    const float qx = qp[0], qy = qp[1], qz = qp[2];
    const bool hiHalf = (lane >= 16);
    v2f Av;
    Av[0] = hiHalf ? qz   : qx;
    Av[1] = hiHalf ? 1.0f : qy;

    // ---- private top-K (sorted ascending; worst at [K-1]) ----
    float dist16[KNN_K];
    int   idx16[KNN_K];
#pragma unroll
    for (int i = 0; i < KNN_K; ++i) { dist16[i] = FLT_BIG; idx16[i] = 0; }

    const float* pbBase = xyz + (size_t)b * NPTS * 3;

    for (int chunk = 0; chunk < NPTS / CHUNK; ++chunk) {
        const int nchunk = chunk * CHUNK;

        // ---- Phase A: 2 WMMAs per wave -> LDS dist[n_local][q] ----
#pragma unroll
        for (int i = 0; i < 2; ++i) {
            const int tw   = wv * 2 + i;          // point tile 0..15 in chunk
            const int nloc = tw * 16 + lm;        // 0..255
            const float* pp = pbBase + (size_t)(nchunk + nloc) * 3;
            const float px = pp[0], py = pp[1], pz = pp[2];
            const float sq = fmaf(px, px, fmaf(py, py, pz * pz));
            // B is 4x16 (KxN): lanes 0-15 supply K0,K1; lanes 16-31 supply K2,K3.
            // Col n: (-2px, -2py, -2pz, ||p||^2)
            v2f Bv;
            Bv[0] = hiHalf ? (-2.0f * pz) : (-2.0f * px);
            Bv[1] = hiHalf ? sq           : (-2.0f * py);

            v8f acc = {};
            acc = __builtin_amdgcn_wmma_f32_16x16x4_f32(
                false, Av, false, Bv, (short)0, acc, false, false);

            // lane<16 holds rows 0-7 of its column; lane>=16 holds rows 8-15
            const int off = nloc * STR + (hiHalf ? 8 : 0);
            *(float4*)&ldsD[off]     = make_float4(acc[0], acc[1], acc[2], acc[3]);
            *(float4*)&ldsD[off + 4] = make_float4(acc[4], acc[5], acc[6], acc[7]);
        }
        __syncthreads();

        // ---- Phase B: each thread scans 16 candidates for its query ----
#pragma unroll
        for (int j = 0; j < 16; ++j) {
            const int nloc = slot + (j << 4);
            const float d  = ldsD[nloc * STR + qid];
            const int gidx = nchunk + nloc;
            if (d < dist16[KNN_K - 1]) {
                float dd = d; int ii = gidx;
#pragma unroll
                for (int t = 0; t < KNN_K; ++t) {
                    const bool sw = dd < dist16[t];
                    const float td = dist16[t]; const int ti = idx16[t];
                    dist16[t] = sw ? dd : td;  idx16[t] = sw ? ii : ti;
                    dd        = sw ? td : dd;  ii       = sw ? ti : ii;
                }
            }
        }
        __syncthreads();
    }

    // ---- dump private lists (already sorted ascending) to LDS ----
#pragma unroll
    for (int k = 0; k < KNN_K; ++k) {
        const int base = (qid * 16 + slot) * KNN_K;
        mD[base + k] = dist16[k];
        mI[base + k] = idx16[k];
    }
    __syncthreads();

    // ---- 16-way streaming merge: one thread per query ----
    if (tid < 16) {
        const int q = tid;
        float hv[16]; int hidx[16]; int pos[16];
#pragma unroll
        for (int s = 0; s < 16; ++s) {
            const int base = (q * 16 + s) * KNN_K;
            hv[s] = mD[base]; hidx[s] = mI[base]; pos[s] = 0;
        }
        const size_t outBase = ((size_t)b * MQRY + (size_t)mtile * 16 + q) * KNN_K;
        for (int k = 0; k < KNN_K; ++k) {
            float bv = hv[0]; int bi = hidx[0]; int bs = 0;
#pragma unroll
            for (int s = 1; s < 16; ++s) {
                const bool better = (hv[s] < bv) || (hv[s] == bv && hidx[s] < bi);
                if (better) { bv = hv[s]; bi = hidx[s]; bs = s; }
            }
            out[outBase + k] = bi;
#pragma unroll
            for (int s = 0; s < 16; ++s) {
                if (s == bs) {
                    pos[s]++;
                    if (pos[s] < KNN_K) {
                        const int base = (q * 16 + s) * KNN_K + pos[s];
                        hv[s] = mD[base]; hidx[s] = mI[base];
                    } else {
                        hv[s] = FLT_BIG;
                    }
                }
            }
        }
    }
}

extern "C" void kernel_launch(void* const* d_in, const int* in_sizes, int n_in,
                              void* d_out, int out_size, void* d_ws, size_t ws_size,
                              hipStream_t stream) {
    const float* xyz     = (const float*)d_in[0];  // [4,16384,3]
    const float* new_xyz = (const float*)d_in[1];  // [4,4096,3]
    int* out = (int*)d_out;                        // [4,4096,16] int32 indices

    const int grid = BATCH * (MQRY / 16);          // 1024 workgroups
    KNN_6030134083767_kernel<<<grid, 256, 0, stream>>>(xyz, new_xyz, out);
}